// PolicyGradientLossFairness_28260884807717
// MI455X (gfx1250) — compile-verified
//
#include <hip/hip_runtime.h>
#include <hip/hip_bf16.h>

#define B_DIM  512
#define M_DIM  1024
#define E_DIM  8
#define MC_DIM 25
#define K_VAL  5

typedef __attribute__((ext_vector_type(2))) float v2f;
typedef __attribute__((ext_vector_type(8))) float v8f;

// One workgroup (256 threads = 8 wave32) per batch row.
__global__ __launch_bounds__(256) void pg_loss_main(
    const float* __restrict__ score,
    const float* __restrict__ relevance,
    const float* __restrict__ eth_list,
    const int*   __restrict__ rank_samples,
    const int*   __restrict__ perm_idx,
    float*       __restrict__ partial)
{
    __shared__ float sdat[M_DIM];   // holds exp(x-max), then overwritten with softmax s
    __shared__ float red[256];
    __shared__ float wsum[8];
    __shared__ float bcast[2];
    __shared__ float mc_part[32];
    // Async-staged index rows (rank rows padded to 32B so b128 dest is 16B-aligned).
    __shared__ __align__(16) int stag_rank[MC_DIM][8];
    __shared__ __align__(16) int stag_perm[MC_DIM][8];

    const int tid  = threadIdx.x;
    const int b    = blockIdx.x;
    const int lane = tid & 31;
    const int wave = tid >> 5;

    // ---- issue async global->LDS copies of the first-K index rows NOW; they
    //      complete in the background while the softmax/WMMA phase runs. ----
    if (tid < MC_DIM) {
        const unsigned rank_off = (unsigned)((b * MC_DIM + tid) * M_DIM) * 4u;
        const unsigned perm_off = (unsigned)((b * MC_DIM + tid) * K_VAL) * 4u;
        const unsigned lds_rank = (unsigned)(uintptr_t)&stag_rank[tid][0];
        const unsigned lds_perm = (unsigned)(uintptr_t)&stag_perm[tid][0];
        // rank_samples first-K: 16B + 4B per MC sample (rows are 4KB apart -> 16B aligned)
        asm volatile("global_load_async_to_lds_b128 %0, %1, %2"
                     :: "v"(lds_rank), "v"(rank_off), "s"(rank_samples) : "memory");
        asm volatile("global_load_async_to_lds_b32 %0, %1, %2 offset:16"
                     :: "v"(lds_rank), "v"(rank_off), "s"(rank_samples) : "memory");
        // perm rows are 20B apart (not 16B aligned) -> five b32 transfers;
        // offset: applies to both the global and LDS address (ISA 08 pseudocode).
        #pragma unroll
        for (int k = 0; k < K_VAL; ++k) {
            asm volatile("global_load_async_to_lds_b32 %0, %1, %2 offset:%3"
                         :: "v"(lds_perm), "v"(perm_off), "s"(perm_idx),
                            "i"(k * 4) : "memory");
        }
    }

    // ---- warm the gather targets for the MC phase (one line per lane). ----
    __builtin_prefetch(relevance + (size_t)b * M_DIM + (size_t)(tid & 63) * 16, 0, 3);
    __builtin_prefetch(eth_list + (size_t)b * M_DIM * E_DIM + (size_t)tid * 16, 0, 3);
    __builtin_prefetch(eth_list + (size_t)b * M_DIM * E_DIM + 4096 + (size_t)tid * 16, 0, 3);

    // ---- load 4 scores per thread (coalesced float4) ----
    const float4 xv = *(const float4*)(score + (size_t)b * M_DIM + tid * 4);

    // ---- row max (deterministic LDS tree) ----
    red[tid] = fmaxf(fmaxf(xv.x, xv.y), fmaxf(xv.z, xv.w));
    __syncthreads();
    #pragma unroll
    for (int s = 128; s > 0; s >>= 1) {
        if (tid < s) red[tid] = fmaxf(red[tid], red[tid + s]);
        __syncthreads();
    }
    const float rmax = red[0];

    // ---- exponentials -> LDS ----
    const float e0 = __expf(xv.x - rmax);
    const float e1 = __expf(xv.y - rmax);
    const float e2 = __expf(xv.z - rmax);
    const float e3 = __expf(xv.w - rmax);
    sdat[tid * 4 + 0] = e0;
    sdat[tid * 4 + 1] = e1;
    sdat[tid * 4 + 2] = e2;
    sdat[tid * 4 + 3] = e3;
    __syncthreads();

    // ---- sum of exponentials via V_WMMA_F32_16X16X4_F32 ----
    // Each wave reduces 128 values: two chained 16x4(A) x 4x16(B=ones) WMMAs.
    // A layout (ISA 7.12.2): lane L<16 -> {A[L][0], A[L][1]}, lane L+16 -> {A[L][2], A[L][3]}
    // with A[m][k] = chunk[m*4+k].  With B = all ones the result rows are exact f32 row sums.
    v2f bones = {1.0f, 1.0f};
    v8f acc = {};
    #pragma unroll
    for (int t = 0; t < 2; ++t) {
        const int c  = wave * 128 + t * 64;
        const int a0 = c + 4 * (lane & 15) + ((lane >> 4) << 1);
        v2f av;
        av.x = sdat[a0];
        av.y = sdat[a0 + 1];
        acc = __builtin_amdgcn_wmma_f32_16x16x4_f32(
            /*neg_a=*/false, av, /*neg_b=*/false, bones,
            /*c_mod=*/(short)0, acc, /*reuse_a=*/false, /*reuse_b=*/false);
    }
    // D layout: VGPR r, lanes 0-15 -> M=r, lanes 16-31 -> M=8+r; every column identical.
    float loc = ((acc[0] + acc[1]) + (acc[2] + acc[3])) +
                ((acc[4] + acc[5]) + (acc[6] + acc[7]));
    loc += __shfl_xor(loc, 16, 32);          // combine M=0..7 with M=8..15 halves
    if (lane == 0) wsum[wave] = loc;
    __syncthreads();
    if (tid == 0) {
        float Z = 0.0f;
        #pragma unroll
        for (int w = 0; w < 8; ++w) Z += wsum[w];
        bcast[1] = Z;
    }
    __syncthreads();
    const float Z = bcast[1];

    // ---- softmax values back to LDS, and T = sum(s) (matches denom[0] up to fp order) ----
    const float s0 = e0 / Z, s1 = e1 / Z, s2 = e2 / Z, s3 = e3 / Z;
    sdat[tid * 4 + 0] = s0;
    sdat[tid * 4 + 1] = s1;
    sdat[tid * 4 + 2] = s2;
    sdat[tid * 4 + 3] = s3;
    red[tid] = (s0 + s1) + (s2 + s3);
    __syncthreads();
    #pragma unroll
    for (int s = 128; s > 0; s >>= 1) {
        if (tid < s) red[tid] += red[tid + s];
        __syncthreads();
    }
    const float T = red[0];

    // ---- Monte-Carlo phase: one lane per sample ----
    if (tid < MC_DIM) {
        // Drain the async copies issued at kernel entry (same wave issued them).
        asm volatile("s_wait_asynccnt 0" ::: "memory");

        int idxk[K_VAL], pk[K_VAL];
        #pragma unroll
        for (int k = 0; k < K_VAL; ++k) idxk[k] = stag_rank[tid][k];
        #pragma unroll
        for (int k = 0; k < K_VAL; ++k) pk[k] = stag_perm[tid][k];

        float fk[K_VAL];
        #pragma unroll
        for (int k = 0; k < K_VAL; ++k) fk[k] = sdat[idxk[k]];

        float rts[K_VAL];
        #pragma unroll
        for (int k = 0; k < K_VAL; ++k) rts[k] = fk[pk[k]];

        // Plackett-Luce: denom[i] = T - sum_{j<i} rts[j]  (reverse-cumsum identity)
        float denom = T, prob = 1.0f;
        #pragma unroll
        for (int k = 0; k < K_VAL; ++k) { prob *= rts[k] / denom; denom -= rts[k]; }
        const float log_prob = __logf(120.0f * prob);   // 5! = 120

        // normalized relevance of top-K (NaN -> 1/K)
        float rel[K_VAL], rsum = 0.0f;
        #pragma unroll
        for (int k = 0; k < K_VAL; ++k) {
            rel[k] = relevance[(size_t)b * M_DIM + idxk[k]];
            rsum  += rel[k];
        }

        // delta_f[e] = sum_k eth[idx_k][e] * rel_k
        float df[E_DIM];
        #pragma unroll
        for (int e = 0; e < E_DIM; ++e) df[e] = 0.0f;
        #pragma unroll
        for (int k = 0; k < K_VAL; ++k) {
            float rk = rel[k] / rsum;
            if (rk != rk) rk = 1.0f / (float)K_VAL;    // NaN guard
            const float4* ep = (const float4*)(eth_list +
                               ((size_t)b * M_DIM + idxk[k]) * E_DIM);
            const float4 ea = ep[0], eb = ep[1];
            df[0] += ea.x * rk; df[1] += ea.y * rk; df[2] += ea.z * rk; df[3] += ea.w * rk;
            df[4] += eb.x * rk; df[5] += eb.y * rk; df[6] += eb.z * rk; df[7] += eb.w * rk;
        }
        const float sdf = ((df[0] + df[1]) + (df[2] + df[3])) +
                          ((df[4] + df[5]) + (df[6] + df[7]));
        float reward = 0.0f;
        #pragma unroll
        for (int e = 0; e < E_DIM; ++e) {
            const float p = df[e] / sdf;
            if (p > 0.0f) reward -= p * __logf(p);     // NaN(p) fails p>0, matches where()
        }
        mc_part[tid] = log_prob * reward;
    }
    __syncthreads();
    if (tid == 0) {
        float a = 0.0f;
        #pragma unroll
        for (int i = 0; i < MC_DIM; ++i) a += mc_part[i];
        partial[b] = a;
    }
}

// Deterministic final reduction over the 512 per-row partials.
__global__ __launch_bounds__(512) void pg_loss_reduce(
    const float* __restrict__ partial, float* __restrict__ out)
{
    __shared__ float red[512];
    const int tid = threadIdx.x;
    red[tid] = partial[tid];
    __syncthreads();
    #pragma unroll
    for (int s = 256; s > 0; s >>= 1) {
        if (tid < s) red[tid] += red[tid + s];
        __syncthreads();
    }
    if (tid == 0) out[0] = -red[0] / (float)(B_DIM * MC_DIM);
}

extern "C" void kernel_launch(void* const* d_in, const int* in_sizes, int n_in,
                              void* d_out, int out_size, void* d_ws, size_t ws_size,
                              hipStream_t stream) {
    const float* score        = (const float*)d_in[0];
    const float* relevance    = (const float*)d_in[1];
    const float* eth_list     = (const float*)d_in[2];
    const int*   rank_samples = (const int*)d_in[3];
    const int*   perm_idx     = (const int*)d_in[4];
    // d_in[5] is K == 5, baked into K_VAL.

    float* partial = (float*)d_ws;   // 512 floats of scratch

    pg_loss_main<<<B_DIM, 256, 0, stream>>>(score, relevance, eth_list,
                                            rank_samples, perm_idx, partial);
    pg_loss_reduce<<<1, B_DIM, 0, stream>>>(partial, (float*)d_out);
}